// QFLayer_23527830847745
// MI455X (gfx1250) — compile-verified
//
#include <hip/hip_runtime.h>
#include <hip/hip_bf16.h>
#include <math.h>

// Reduced computation (cross-attn K/V are broadcast of im_repr -> softmax is
// exactly uniform -> ctx == V; the whole self-attn / Q,K cross path is dead):
//   x = im_repr @ Wv2 + bv2                      [256,1024]x[1024,768]
//   u = gelu_exact(x @ W1 + b1)                  [256,768]x[768,3072]
//   y = x + (u @ W2 + b2)                        [256,3072]x[3072,768]
//   out = LayerNorm(y; ln_g, ln_b, eps=1e-12)
//
// ~2.8 GFLOP / ~23 MB compulsory HBM traffic -> memory-roofline (~1us at
// 23.3 TB/s). Full f32 WMMA keeps bit-level accuracy; B panels are staged to
// LDS by the Tensor Data Mover (double-buffered, overlapped with WMMA).

#define NB   256
#define DMOD 768
#define REPR 1024
#define FFW  3072
#define KC   32      // K-chunk staged per TDM transfer (32 x 64 floats = 8 KB)

typedef __attribute__((ext_vector_type(2))) float v2f;
typedef __attribute__((ext_vector_type(8))) float v8f;
typedef __attribute__((ext_vector_type(4))) unsigned int v4u;
typedef __attribute__((ext_vector_type(4))) int v4i;
typedef __attribute__((ext_vector_type(8))) int v8i;

// Issue one 2-D TDM tile load: rows x cols elements of f32, row stride `ld`
// elements, from `gaddr` into LDS byte offset `lds_off`.
// D# packing per CDNA5 ISA 8.3/8.4 (group0: count/lds/global/type,
// group1: data_size=4B, tensor dims, tile dims, dim0 stride).
__device__ __forceinline__ void tdm_load_2d(unsigned lds_off, const float* gaddr,
                                            unsigned rows, unsigned cols,
                                            unsigned ld, unsigned tdim0,
                                            unsigned tdim1) {
  unsigned long long ga = (unsigned long long)(uintptr_t)gaddr;
  v4u g0;
  g0.x = 1u;                                               // count=1 (valid user D#)
  g0.y = lds_off;                                          // lds_addr (bytes)
  g0.z = (unsigned)(ga & 0xFFFFFFFFu);                     // global_addr[31:0]
  g0.w = (unsigned)((ga >> 32) & 0x1FFFFFFu) | (2u << 30); // global_addr[56:32] | type=2
  v8i g1;
  g1[0] = (int)(2u << 16);                                  // data_size=2 (4 bytes)
  g1[1] = (int)((tdim0 & 0xFFFFu) << 16);                   // tensor_dim0[15:0] @bits63:48
  g1[2] = (int)(((tdim0 >> 16) & 0xFFFFu) | ((tdim1 & 0xFFFFu) << 16));
  g1[3] = (int)(((tdim1 >> 16) & 0xFFFFu) | ((cols & 0xFFFFu) << 16)); // tile_dim0
  g1[4] = (int)(rows & 0xFFFFu);                            // tile_dim1 (tile_dim2=0)
  g1[5] = (int)ld;                                          // tensor_dim0_stride[31:0]
  g1[6] = 0;                                                // stride0 hi / stride1 lo
  g1[7] = 0;
  v4i z4 = {0, 0, 0, 0};
#if __clang_major__ >= 23
  v8i z8 = {0, 0, 0, 0, 0, 0, 0, 0};
  __builtin_amdgcn_tensor_load_to_lds(g0, g1, z4, z4, z8, 0);
#else
  __builtin_amdgcn_tensor_load_to_lds(g0, g1, z4, z4, 0);
#endif
}

// Block = 4 waves = 16(M) x 64(N) output. Each wave32 owns one 16x16 tile and
// steps K by 4 with V_WMMA_F32_16X16X4_F32. B panel (KC x 64) comes from LDS,
// DMA'd by the TDM with double buffering; TDM of chunk c+1 overlaps compute
// of chunk c. A fragments are contiguous global b64 loads (shared by all 4
// waves via WGP$/L2).
template <int MODE>  // 0: +bias   1: +bias,gelu   2: +bias,+residual
__global__ __launch_bounds__(128) void gemm_wmma_tdm(
    const float* __restrict__ A, const float* __restrict__ B,
    const float* __restrict__ bias, const float* __restrict__ resid,
    float* __restrict__ out, int N, int K)
{
  __shared__ float bpanel[2][KC * 64];   // 2 x 8 KB

  const int lane = threadIdx.x & 31;
  const int wave = threadIdx.x >> 5;
  const int nblk = N >> 6;                       // 64-column blocks
  const int m0   = (blockIdx.x / nblk) * 16;
  const int nb0  = (blockIdx.x % nblk) * 64;
  const int half = lane >> 4;                    // which K-pair this half-wave holds
  const int idx  = lane & 15;                    // row (A) / col (B) in tile

  const float* arow = A + (size_t)(m0 + idx) * K + half * 2;
  const unsigned lds0 = (unsigned)(uintptr_t)&bpanel[0][0];
  const unsigned lds1 = (unsigned)(uintptr_t)&bpanel[1][0];

  const int C = K / KC;

  if (wave == 0) {
    tdm_load_2d(lds0, B + nb0, KC, 64, (unsigned)N, (unsigned)N, (unsigned)K);
    __builtin_amdgcn_s_wait_tensorcnt(0);
  }
  __syncthreads();

  v8f acc = {0.f, 0.f, 0.f, 0.f, 0.f, 0.f, 0.f, 0.f};

  for (int c = 0; c < C; ++c) {
    // Prefetch next B panel into the other buffer (overlaps WMMA below).
    if (wave == 0 && (c + 1) < C) {
      tdm_load_2d((c & 1) ? lds0 : lds1,
                  B + (size_t)(c + 1) * KC * N + nb0,
                  KC, 64, (unsigned)N, (unsigned)N, (unsigned)K);
    }

    const float* bp = &bpanel[c & 1][0] + half * 2 * 64 + wave * 16 + idx;
#pragma unroll
    for (int kk = 0; kk < KC; kk += 4) {
      v2f a = *(const v2f*)(arow + kk);       // A[m][k+2h .. k+2h+1]
      v2f b;
      b.x = bp[kk * 64];                      // B[k+2h  ][n] from LDS
      b.y = bp[kk * 64 + 64];                 // B[k+2h+1][n] from LDS
      acc = __builtin_amdgcn_wmma_f32_16x16x4_f32(
          false, a, false, b, (short)0, acc, false, false);
    }
    arow += KC;

    if (wave == 0) __builtin_amdgcn_s_wait_tensorcnt(0);  // next panel landed
    __syncthreads();                                      // safe to swap buffers
  }

  const int n  = nb0 + wave * 16 + idx;
  const float bn = bias[n];
#pragma unroll
  for (int r = 0; r < 8; ++r) {
    const int m = m0 + r + half * 8;
    float v = acc[r] + bn;
    if (MODE == 1) v = 0.5f * v * (1.0f + erff(v * 0.70710678118654752f));  // exact GELU
    if (MODE == 2) v += resid[(size_t)m * N + n];
    out[(size_t)m * N + n] = v;
  }
}

// Row-wise LayerNorm over 768 columns, one block per row.
__global__ __launch_bounds__(256) void layernorm_kernel(
    const float* __restrict__ y, const float* __restrict__ g,
    const float* __restrict__ b, float* __restrict__ out)
{
  __shared__ float red[256];
  const int row = blockIdx.x;
  const int tid = threadIdx.x;
  const float* yr = y + (size_t)row * DMOD;

  float s = 0.f;
  for (int i = tid; i < DMOD; i += 256) s += yr[i];
  red[tid] = s; __syncthreads();
  for (int off = 128; off > 0; off >>= 1) {
    if (tid < off) red[tid] += red[tid + off];
    __syncthreads();
  }
  const float mu = red[0] * (1.0f / DMOD);
  __syncthreads();

  float v = 0.f;
  for (int i = tid; i < DMOD; i += 256) { float d = yr[i] - mu; v += d * d; }
  red[tid] = v; __syncthreads();
  for (int off = 128; off > 0; off >>= 1) {
    if (tid < off) red[tid] += red[tid + off];
    __syncthreads();
  }
  const float rstd = rsqrtf(red[0] * (1.0f / DMOD) + 1e-12f);

  for (int i = tid; i < DMOD; i += 256)
    out[(size_t)row * DMOD + i] = (yr[i] - mu) * rstd * g[i] + b[i];
}

extern "C" void kernel_launch(void* const* d_in, const int* in_sizes, int n_in,
                              void* d_out, int out_size, void* d_ws, size_t ws_size,
                              hipStream_t stream) {
  (void)in_sizes; (void)n_in; (void)out_size; (void)ws_size;

  const float* im  = (const float*)d_in[0];   // [256,1024]
  const float* Wv2 = (const float*)d_in[12];  // [1024,768]
  const float* bv2 = (const float*)d_in[13];  // [768]
  const float* W1  = (const float*)d_in[14];  // [768,3072]
  const float* b1  = (const float*)d_in[15];  // [3072]
  const float* W2  = (const float*)d_in[16];  // [3072,768]
  const float* b2  = (const float*)d_in[17];  // [768]
  const float* lng = (const float*)d_in[18];  // [768]
  const float* lnb = (const float*)d_in[19];  // [768]

  float* x = (float*)d_ws;          // [256,768]
  float* u = x + NB * DMOD;         // [256,3072]
  float* y = u + NB * FFW;          // [256,768]

  // x = im @ Wv2 + bv2 : 16 x 12 blocks of 16x64
  gemm_wmma_tdm<0><<<16 * (DMOD / 64), 128, 0, stream>>>(im, Wv2, bv2, nullptr, x, DMOD, REPR);
  // u = gelu(x @ W1 + b1)
  gemm_wmma_tdm<1><<<16 * (FFW / 64), 128, 0, stream>>>(x, W1, b1, nullptr, u, FFW, DMOD);
  // y = x + (u @ W2 + b2)
  gemm_wmma_tdm<2><<<16 * (DMOD / 64), 128, 0, stream>>>(u, W2, b2, x, y, DMOD, FFW);
  // out = LayerNorm(y)
  layernorm_kernel<<<NB, 256, 0, stream>>>(y, lng, lnb, (float*)d_out);
}